// SimpleGGNN_59425167507918
// MI455X (gfx1250) — compile-verified
//
#include <hip/hip_runtime.h>
#include <hip/hip_bf16.h>

// ---------------- problem constants (match reference) ----------------
#define N_NODES  100000
#define D_IN     128
#define HID      64
#define N_LAYERS 5
#define D_OUT    32
#define N_GRAPHS 1000
#define G_PAD    1008          // N_GRAPHS padded up to multiple of 16

typedef __attribute__((ext_vector_type(2))) float v2f;
typedef __attribute__((ext_vector_type(8))) float v8f;

// fp32 WMMA: D(16x16,f32) = A(16x4,f32) * B(4x16,f32) + C
__device__ __forceinline__ v8f wmma_f32(v2f a, v2f b, v8f c) {
    // (neg_a, A, neg_b, B, c_mod, C, reuse_a, reuse_b)
    return __builtin_amdgcn_wmma_f32_16x16x4_f32(false, a, false, b,
                                                 (short)0, c, false, false);
}

// ---------------------------------------------------------------------
// Generic row-major GEMM:  C[M,N] = A[M,K] @ B[K,N] (+ bias[N])
// One wave per 16x16 output tile. M multiple of 16; stores guarded by Mstore.
// fp32 WMMA VGPR layout (wave32):
//   A 16x4 : lane&15 = M row ; K = k0 + 2*(lane>>4) + {0,1} in v0/v1
//   B 4x16 : lane&15 = N col ; same K split
//   C 16x16: VGPR v, lane<16 -> M=v, N=lane ; lane>=16 -> M=v+8, N=lane-16
// ---------------------------------------------------------------------
template <int K, int N>
__global__ void wmma_gemm_bias(const float* __restrict__ A,
                               const float* __restrict__ B,
                               const float* __restrict__ bias,
                               float* __restrict__ C,
                               int M, int Mstore) {
    const int wave = (blockIdx.x * blockDim.x + threadIdx.x) >> 5;
    const int lane = threadIdx.x & 31;
    const int ntile = N / 16;
    const int T = (M / 16) * ntile;
    if (wave >= T) return;

    const int rowBase = (wave / ntile) * 16;
    const int colBase = (wave % ntile) * 16;
    const int half = lane >> 4;
    const int r    = lane & 15;

    v8f acc = {};
    const float* arow = A + (size_t)(rowBase + r) * K + 2 * half;
    const int col = colBase + r;

#pragma unroll 4
    for (int k0 = 0; k0 < K; k0 += 4) {
        v2f a = *(const v2f*)(arow + k0);          // A[row][k0+2h], +1
        const int k = k0 + 2 * half;
        v2f b;
        b.x = B[(size_t)k * N + col];              // B[k][col]
        b.y = B[(size_t)(k + 1) * N + col];        // B[k+1][col]
        acc = wmma_f32(a, b, acc);
    }

    const float bv = bias ? bias[col] : 0.0f;
#pragma unroll
    for (int v = 0; v < 8; ++v) {
        const int row = rowBase + v + 8 * half;
        if (row < Mstore)
            C[(size_t)row * N + col] = acc[v] + bv;
    }
}

// ---------------------------------------------------------------------
// Fused GRU cell over one 16x16 (node x hidden) tile.
//   gi = agg @ w_ih^T + b_ih ; gh = h @ w_hh^T + b_hh   (w_* : [3H, H])
//   r = sig(i_r+h_r); z = sig(i_z+h_z); n = tanh(i_n + r*h_n)
//   h' = (1-z)*n + z*h
// B matrices are w^T, so B[k][j] = w[j*H + k]  -> contiguous float2 in k.
// ---------------------------------------------------------------------
__global__ void gru_fused(const float* __restrict__ agg,
                          const float* __restrict__ h,
                          const float* __restrict__ w_ih,
                          const float* __restrict__ w_hh,
                          const float* __restrict__ b_ih,
                          const float* __restrict__ b_hh,
                          float* __restrict__ h_out) {
    const int wave = (blockIdx.x * blockDim.x + threadIdx.x) >> 5;
    const int lane = threadIdx.x & 31;
    const int ntile = HID / 16;                     // 4
    const int T = (N_NODES / 16) * ntile;           // 25000
    if (wave >= T) return;

    const int rowBase = (wave / ntile) * 16;
    const int colBase = (wave % ntile) * 16;
    const int half = lane >> 4;
    const int r    = lane & 15;
    const int col  = colBase + r;

    v8f ai[3] = {{}, {}, {}};
    v8f ah[3] = {{}, {}, {}};

    const float* aggrow = agg  + (size_t)(rowBase + r) * HID + 2 * half;
    const float* hrow   = h    + (size_t)(rowBase + r) * HID + 2 * half;
    const float* wi0    = w_ih + (size_t)col * HID + 2 * half;   // gate g: + g*H*H
    const float* wh0    = w_hh + (size_t)col * HID + 2 * half;

#pragma unroll 4
    for (int k0 = 0; k0 < HID; k0 += 4) {
        v2f a0 = *(const v2f*)(aggrow + k0);
        v2f a1 = *(const v2f*)(hrow + k0);
#pragma unroll
        for (int g = 0; g < 3; ++g) {
            v2f bi = *(const v2f*)(wi0 + (size_t)g * HID * HID + k0);
            v2f bh = *(const v2f*)(wh0 + (size_t)g * HID * HID + k0);
            ai[g] = wmma_f32(a0, bi, ai[g]);
            ah[g] = wmma_f32(a1, bh, ah[g]);
        }
    }

    const float bir = b_ih[col], biz = b_ih[HID + col], bin_ = b_ih[2 * HID + col];
    const float bhr = b_hh[col], bhz = b_hh[HID + col], bhn  = b_hh[2 * HID + col];

#pragma unroll
    for (int v = 0; v < 8; ++v) {
        const int row = rowBase + v + 8 * half;
        const float hold = h[(size_t)row * HID + col];
        const float rr = 1.0f / (1.0f + __expf(-((ai[0][v] + bir) + (ah[0][v] + bhr))));
        const float zz = 1.0f / (1.0f + __expf(-((ai[1][v] + biz) + (ah[1][v] + bhz))));
        const float nn = tanhf((ai[2][v] + bin_) + rr * (ah[2][v] + bhn));
        h_out[(size_t)row * HID + col] = (1.0f - zz) * nn + zz * hold;
    }
}

// ---------------------------------------------------------------------
// Edge scatter-add: agg[dst[e]] += m[src[e]].  16 lanes (4 floats each)
// per edge -> coalesced float4 gather + grouped f32 atomics.
// ---------------------------------------------------------------------
__global__ void scatter_add(const float* __restrict__ m,
                            const int* __restrict__ src,
                            const int* __restrict__ dst,
                            float* __restrict__ agg, int n_edges) {
    const int t = blockIdx.x * blockDim.x + threadIdx.x;
    const int e = t >> 4;
    if (e >= n_edges) return;
    const int part = t & 15;
    const int s = src[e], d = dst[e];
    const float4 val = *(const float4*)(m + (size_t)s * HID + part * 4);
    float* out = agg + (size_t)d * HID + part * 4;
    atomicAdd(out + 0, val.x);
    atomicAdd(out + 1, val.y);
    atomicAdd(out + 2, val.z);
    atomicAdd(out + 3, val.w);
}

// Mean-pool accumulation: sums[batch[n]] += h[n]; cnt[batch[n]] += 1
__global__ void pool_add(const float* __restrict__ h,
                         const int* __restrict__ batch,
                         float* __restrict__ sums,
                         float* __restrict__ cnt) {
    const int t = blockIdx.x * blockDim.x + threadIdx.x;
    const int n = t >> 4;
    if (n >= N_NODES) return;
    const int part = t & 15;
    const int g = batch[n];
    const float4 val = *(const float4*)(h + (size_t)n * HID + part * 4);
    float* out = sums + (size_t)g * HID + part * 4;
    atomicAdd(out + 0, val.x);
    atomicAdd(out + 1, val.y);
    atomicAdd(out + 2, val.z);
    atomicAdd(out + 3, val.w);
    if (part == 0) atomicAdd(cnt + g, 1.0f);
}

// pooled[g] = sums[g] / max(cnt[g],1)  (rows >= N_GRAPHS zero-padded)
__global__ void pool_finalize(const float* __restrict__ sums,
                              const float* __restrict__ cnt,
                              float* __restrict__ pooled) {
    const int t = blockIdx.x * blockDim.x + threadIdx.x;
    if (t >= G_PAD * HID) return;
    const int g = t / HID;
    float v = 0.0f;
    if (g < N_GRAPHS) v = sums[t] / fmaxf(cnt[g], 1.0f);
    pooled[t] = v;
}

__global__ void zero_f4(float4* __restrict__ p, int n4) {
    const int t = blockIdx.x * blockDim.x + threadIdx.x;
    if (t < n4) p[t] = make_float4(0.f, 0.f, 0.f, 0.f);
}

// ---------------------------------------------------------------------
extern "C" void kernel_launch(void* const* d_in, const int* in_sizes, int n_in,
                              void* d_out, int out_size, void* d_ws, size_t ws_size,
                              hipStream_t stream) {
    const float* node_embed = (const float*)d_in[0];
    const int*   edge_index = (const int*)  d_in[1];
    const int*   batch      = (const int*)  d_in[2];
    const float* W_in       = (const float*)d_in[3];
    const float* b_in       = (const float*)d_in[4];
    const float* ggnn_w     = (const float*)d_in[5];
    const float* gru_w_ih   = (const float*)d_in[6];
    const float* gru_w_hh   = (const float*)d_in[7];
    const float* gru_b_ih   = (const float*)d_in[8];
    const float* gru_b_hh   = (const float*)d_in[9];
    const float* W_out      = (const float*)d_in[10];
    const float* b_out      = (const float*)d_in[11];

    const int n_edges = in_sizes[1] / 2;
    const int* src = edge_index;            // edge_index[0, :]
    const int* dst = edge_index + n_edges;  // edge_index[1, :]

    // ---- carve workspace ----
    char* ws = (char*)d_ws;
    const size_t node_bytes = (size_t)N_NODES * HID * sizeof(float);  // 25.6 MB
    float* bufA = (float*)ws;              ws += node_bytes;   // h
    float* bufB = (float*)ws;              ws += node_bytes;   // m / h_new (ping-pong)
    float* agg  = (float*)ws;              ws += node_bytes;
    float* sums = (float*)ws;              ws += (size_t)G_PAD * HID * sizeof(float);
    float* cnt  = (float*)ws;              ws += (size_t)G_PAD * sizeof(float);
    float* pooled = (float*)ws;            ws += (size_t)G_PAD * HID * sizeof(float);

    const int TPB = 256;
    const int tiles_nodes = (N_NODES / 16) * (HID / 16);        // 25000 waves
    const int gemm_blocks = (tiles_nodes + 7) / 8;              // 8 waves/block

    // 1) h = node_embed @ W_in + b_in
    wmma_gemm_bias<D_IN, HID><<<gemm_blocks, TPB, 0, stream>>>(
        node_embed, W_in, b_in, bufA, N_NODES, N_NODES);

    // 2) GGNN layers
    float* hcur = bufA;
    float* hnxt = bufB;
    const int agg_f4 = N_NODES * HID / 4;
    for (int l = 0; l < N_LAYERS; ++l) {
        // m = h @ w_l
        wmma_gemm_bias<HID, HID><<<gemm_blocks, TPB, 0, stream>>>(
            hcur, ggnn_w + (size_t)l * HID * HID, nullptr, hnxt, N_NODES, N_NODES);

        // agg = 0 ; agg[dst] += m[src]
        zero_f4<<<(agg_f4 + TPB - 1) / TPB, TPB, 0, stream>>>((float4*)agg, agg_f4);
        scatter_add<<<((size_t)n_edges * 16 + TPB - 1) / TPB, TPB, 0, stream>>>(
            hnxt, src, dst, agg, n_edges);

        // h_new = GRU(agg, h)  (overwrites message buffer, then swap)
        gru_fused<<<gemm_blocks, TPB, 0, stream>>>(
            agg, hcur, gru_w_ih, gru_w_hh, gru_b_ih, gru_b_hh, hnxt);

        float* tmp = hcur; hcur = hnxt; hnxt = tmp;
    }

    // 3) global mean pool
    const int sums_f4 = G_PAD * HID / 4;
    const int cnt_f4  = G_PAD / 4;
    zero_f4<<<(sums_f4 + TPB - 1) / TPB, TPB, 0, stream>>>((float4*)sums, sums_f4);
    zero_f4<<<(cnt_f4 + TPB - 1) / TPB, TPB, 0, stream>>>((float4*)cnt, cnt_f4);
    pool_add<<<(N_NODES * 16 + TPB - 1) / TPB, TPB, 0, stream>>>(hcur, batch, sums, cnt);
    pool_finalize<<<(G_PAD * HID + TPB - 1) / TPB, TPB, 0, stream>>>(sums, cnt, pooled);

    // 4) out = pooled @ W_out + b_out   (pad rows are zero; stores guarded)
    const int tiles_out = (G_PAD / 16) * (D_OUT / 16);          // 126 waves
    wmma_gemm_bias<HID, D_OUT><<<(tiles_out + 7) / 8, TPB, 0, stream>>>(
        pooled, W_out, b_out, (float*)d_out, G_PAD, N_GRAPHS);
}